// CAM_71562745086362
// MI455X (gfx1250) — compile-verified
//
#include <hip/hip_runtime.h>

typedef float v2f __attribute__((ext_vector_type(2)));
typedef float v8f __attribute__((ext_vector_type(8)));

#define BATCH 8
#define CCH   512
#define NSP   4096

#define WMMA_F32(A, B, Cacc) \
    __builtin_amdgcn_wmma_f32_16x16x4_f32(false, (A), false, (B), (short)0, (Cacc), false, false)

// ---- CDNA5 async global->LDS path (guarded; falls back cleanly) ----
#if defined(__has_builtin)
#if __has_builtin(__builtin_amdgcn_global_load_async_to_lds_b128) && \
    __has_builtin(__builtin_amdgcn_s_wait_asynccnt)
#define HAVE_ASYNC_LDS 1
#endif
#endif

#if defined(HAVE_ASYNC_LDS)
// Probe-derived signature: (v4i addrspace(1)* src, v4i addrspace(3)* dst, Ii offset, Ii cpol)
typedef int v4i __attribute__((__vector_size__(16)));
typedef __attribute__((address_space(1))) v4i g_v4i;
typedef __attribute__((address_space(3))) v4i l_v4i;
__device__ __forceinline__ void async_copy16(const float* gsrc, float* ldst) {
    __builtin_amdgcn_global_load_async_to_lds_b128(
        (g_v4i*)gsrc, (l_v4i*)ldst, /*offset=*/0, /*cpol=*/0);
}
__device__ __forceinline__ void async_publish() {
    __builtin_amdgcn_s_wait_asynccnt(0);
}
#else
__device__ __forceinline__ void async_publish() {}
#endif

// ---------------------------------------------------------------------------
// Kernel 1: energy[b] = X[b] * X[b]^T   (512 x 512, K = 4096)
// 256 threads (8 waves). Block tile 64(M) x 128(N), K-block 32, double-
// buffered LDS: async DMA of panel k+1 overlaps the 32-WMMA compute of k.
// Wave grid 2x4; each wave computes a 32x32 tile = 2x2 WMMA fragments.
// ---------------------------------------------------------------------------
__global__ __launch_bounds__(256) void cam_energy(const float* __restrict__ x,
                                                  float* __restrict__ energy) {
    __shared__ float As[2][64][33];
    __shared__ float Bs[2][128][33];

    const int b    = blockIdx.z;
    const int m0   = blockIdx.y * 64;
    const int n0   = blockIdx.x * 128;
    const int tid  = threadIdx.x;
    const int lane = tid & 31;
    const int wave = tid >> 5;
    const int wm   = wave >> 2;     // 0..1  -> 32-row slab
    const int wn   = wave & 3;      // 0..3  -> 32-col slab

    const float* xb = x + (size_t)b * CCH * NSP;

    v8f acc00 = {}, acc01 = {}, acc10 = {}, acc11 = {};

    const int lr = tid >> 3;        // 0..31
    const int lk = (tid & 7) * 4;   // 0,4,...,28

    // per-thread staging closure
    auto stage = [&](int buf, int k0) {
#if defined(HAVE_ASYNC_LDS)
        async_copy16(xb + (size_t)(m0 + lr)      * NSP + k0 + lk, &As[buf][lr][lk]);
        async_copy16(xb + (size_t)(m0 + 32 + lr) * NSP + k0 + lk, &As[buf][32 + lr][lk]);
        async_copy16(xb + (size_t)(n0 + lr)      * NSP + k0 + lk, &Bs[buf][lr][lk]);
        async_copy16(xb + (size_t)(n0 + 32 + lr) * NSP + k0 + lk, &Bs[buf][32 + lr][lk]);
        async_copy16(xb + (size_t)(n0 + 64 + lr) * NSP + k0 + lk, &Bs[buf][64 + lr][lk]);
        async_copy16(xb + (size_t)(n0 + 96 + lr) * NSP + k0 + lk, &Bs[buf][96 + lr][lk]);
#else
        const float4 va0 = *(const float4*)(xb + (size_t)(m0 + lr)      * NSP + k0 + lk);
        const float4 va1 = *(const float4*)(xb + (size_t)(m0 + 32 + lr) * NSP + k0 + lk);
        const float4 vb0 = *(const float4*)(xb + (size_t)(n0 + lr)      * NSP + k0 + lk);
        const float4 vb1 = *(const float4*)(xb + (size_t)(n0 + 32 + lr) * NSP + k0 + lk);
        const float4 vb2 = *(const float4*)(xb + (size_t)(n0 + 64 + lr) * NSP + k0 + lk);
        const float4 vb3 = *(const float4*)(xb + (size_t)(n0 + 96 + lr) * NSP + k0 + lk);
        float* a0p = &As[buf][lr][lk];      a0p[0]=va0.x; a0p[1]=va0.y; a0p[2]=va0.z; a0p[3]=va0.w;
        float* a1p = &As[buf][32+lr][lk];   a1p[0]=va1.x; a1p[1]=va1.y; a1p[2]=va1.z; a1p[3]=va1.w;
        float* b0p = &Bs[buf][lr][lk];      b0p[0]=vb0.x; b0p[1]=vb0.y; b0p[2]=vb0.z; b0p[3]=vb0.w;
        float* b1p = &Bs[buf][32+lr][lk];   b1p[0]=vb1.x; b1p[1]=vb1.y; b1p[2]=vb1.z; b1p[3]=vb1.w;
        float* b2p = &Bs[buf][64+lr][lk];   b2p[0]=vb2.x; b2p[1]=vb2.y; b2p[2]=vb2.z; b2p[3]=vb2.w;
        float* b3p = &Bs[buf][96+lr][lk];   b3p[0]=vb3.x; b3p[1]=vb3.y; b3p[2]=vb3.z; b3p[3]=vb3.w;
#endif
    };

    // prologue: fill buffer 0
    stage(0, 0);
    async_publish();
    __syncthreads();

    const int krow = 2 * (lane >> 4);        // lanes 0-15: K+0/1, 16-31: K+2/3
    const int ar   = lane & 15;

    int p = 0;
    for (int k0 = 0; k0 < NSP; k0 += 32) {
        const bool has_next = (k0 + 32 < NSP);
        if (has_next) {
            stage(p ^ 1, k0 + 32);           // async DMA overlaps compute below
            if (k0 + 64 < NSP) {             // prefetch panel k+2 into GL2
                __builtin_prefetch(xb + (size_t)(m0 + lr) * NSP + k0 + 64 + lk, 0, 3);
                __builtin_prefetch(xb + (size_t)(n0 + lr) * NSP + k0 + 64 + lk, 0, 3);
            }
        }

#pragma unroll
        for (int kk = 0; kk < 32; kk += 4) {
            v2f a0, a1, b0, b1;
            a0.x = As[p][wm*32 + ar]     [kk + krow];  a0.y = As[p][wm*32 + ar]     [kk + krow + 1];
            a1.x = As[p][wm*32 + 16 + ar][kk + krow];  a1.y = As[p][wm*32 + 16 + ar][kk + krow + 1];
            b0.x = Bs[p][wn*32 + ar]     [kk + krow];  b0.y = Bs[p][wn*32 + ar]     [kk + krow + 1];
            b1.x = Bs[p][wn*32 + 16 + ar][kk + krow];  b1.y = Bs[p][wn*32 + 16 + ar][kk + krow + 1];
            acc00 = WMMA_F32(a0, b0, acc00);
            acc01 = WMMA_F32(a0, b1, acc01);
            acc10 = WMMA_F32(a1, b0, acc10);
            acc11 = WMMA_F32(a1, b1, acc11);
        }

        if (has_next) {
            async_publish();                  // next buffer fill complete
            __syncthreads();                  // all waves done with buffer p
            p ^= 1;
        }
    }

    // C/D layout: VGPR r -> M=r (lanes 0-15) / M=r+8 (lanes 16-31); N = lane%16
    float* eb = energy + (size_t)b * CCH * CCH;
    const int Mf = m0 + wm*32 + 8 * (lane >> 4);
    const int Nf = n0 + wn*32 + (lane & 15);
#pragma unroll
    for (int r = 0; r < 8; ++r) {
        eb[(size_t)(Mf + r)      * CCH + Nf]      = acc00[r];
        eb[(size_t)(Mf + r)      * CCH + Nf + 16] = acc01[r];
        eb[(size_t)(Mf + 16 + r) * CCH + Nf]      = acc10[r];
        eb[(size_t)(Mf + 16 + r) * CCH + Nf + 16] = acc11[r];
    }
}

// ---------------------------------------------------------------------------
// Kernel 2: inverted softmax per energy row.
// softmax(rowmax - e) == exp(rowmin - e) / sum(exp(rowmin - e))   (exact)
// ---------------------------------------------------------------------------
__global__ __launch_bounds__(256) void cam_softmax(float* __restrict__ energy) {
    __shared__ float red[256];
    float* e = energy + (size_t)blockIdx.x * CCH;
    const int tid = threadIdx.x;

    const float v0 = e[tid];
    const float v1 = e[tid + 256];

    red[tid] = fminf(v0, v1);
    __syncthreads();
#pragma unroll
    for (int s = 128; s > 0; s >>= 1) {
        if (tid < s) red[tid] = fminf(red[tid], red[tid + s]);
        __syncthreads();
    }
    const float mn = red[0];
    __syncthreads();

    const float t0 = expf(mn - v0);
    const float t1 = expf(mn - v1);
    red[tid] = t0 + t1;
    __syncthreads();
#pragma unroll
    for (int s = 128; s > 0; s >>= 1) {
        if (tid < s) red[tid] += red[tid + s];
        __syncthreads();
    }
    const float inv = 1.0f / red[0];

    e[tid]       = t0 * inv;
    e[tid + 256] = t1 * inv;
}

// ---------------------------------------------------------------------------
// Kernel 3: out[b] = gamma * (attn[b] * X[b]) + x[b]
// Block tile 64(M=C) x 128(N=spatial), K = 512 blocked by 32, double-buffered.
// ---------------------------------------------------------------------------
__global__ __launch_bounds__(256) void cam_out(const float* __restrict__ x,
                                               const float* __restrict__ attn,
                                               const float* __restrict__ gamma,
                                               float* __restrict__ out) {
    __shared__ float As[2][64][33];    // attn: 64 M-rows x 32 K (padded)
    __shared__ float Bs[2][32][132];   // X:    32 K-rows x 128 N (padded)

    const int b    = blockIdx.z;
    const int m0   = blockIdx.y * 64;
    const int n0   = blockIdx.x * 128;
    const int tid  = threadIdx.x;
    const int lane = tid & 31;
    const int wave = tid >> 5;
    const int wm   = wave >> 2;
    const int wn   = wave & 3;

    const float* ab = attn + (size_t)b * CCH * CCH;
    const float* xb = x    + (size_t)b * CCH * NSP;

    v8f acc00 = {}, acc01 = {}, acc10 = {}, acc11 = {};

    const int lrA = tid >> 3;          // 0..31
    const int lkA = (tid & 7) * 4;     // 0..28
    const int krB = tid >> 5;          // 0..7
    const int ncB = (tid & 31) * 4;    // 0..124

    auto stage = [&](int buf, int k0) {
#if defined(HAVE_ASYNC_LDS)
        async_copy16(ab + (size_t)(m0 + lrA)      * CCH + k0 + lkA, &As[buf][lrA][lkA]);
        async_copy16(ab + (size_t)(m0 + 32 + lrA) * CCH + k0 + lkA, &As[buf][32 + lrA][lkA]);
#pragma unroll
        for (int q = 0; q < 4; ++q) {
            const int kr = krB + q * 8;
            async_copy16(xb + (size_t)(k0 + kr) * NSP + n0 + ncB, &Bs[buf][kr][ncB]);
        }
#else
        const float4 va0 = *(const float4*)(ab + (size_t)(m0 + lrA)      * CCH + k0 + lkA);
        const float4 va1 = *(const float4*)(ab + (size_t)(m0 + 32 + lrA) * CCH + k0 + lkA);
        float* a0p = &As[buf][lrA][lkA];    a0p[0]=va0.x; a0p[1]=va0.y; a0p[2]=va0.z; a0p[3]=va0.w;
        float* a1p = &As[buf][32+lrA][lkA]; a1p[0]=va1.x; a1p[1]=va1.y; a1p[2]=va1.z; a1p[3]=va1.w;
#pragma unroll
        for (int q = 0; q < 4; ++q) {
            const int kr = krB + q * 8;
            const float4 vb = *(const float4*)(xb + (size_t)(k0 + kr) * NSP + n0 + ncB);
            float* bp = &Bs[buf][kr][ncB];  bp[0]=vb.x; bp[1]=vb.y; bp[2]=vb.z; bp[3]=vb.w;
        }
#endif
    };

    stage(0, 0);
    async_publish();
    __syncthreads();

    const int krow = 2 * (lane >> 4);
    const int ar   = lane & 15;

    int p = 0;
    for (int k0 = 0; k0 < CCH; k0 += 32) {
        const bool has_next = (k0 + 32 < CCH);
        if (has_next) {
            stage(p ^ 1, k0 + 32);
            if (k0 + 64 < CCH) {
                __builtin_prefetch(ab + (size_t)(m0 + lrA) * CCH + k0 + 64 + lkA, 0, 3);
                __builtin_prefetch(xb + (size_t)(k0 + 64 + krB) * NSP + n0 + ncB, 0, 3);
            }
        }

#pragma unroll
        for (int kk = 0; kk < 32; kk += 4) {
            v2f a0, a1, b0, b1;
            a0.x = As[p][wm*32 + ar]     [kk + krow];  a0.y = As[p][wm*32 + ar]     [kk + krow + 1];
            a1.x = As[p][wm*32 + 16 + ar][kk + krow];  a1.y = As[p][wm*32 + 16 + ar][kk + krow + 1];
            b0.x = Bs[p][kk + krow][wn*32 + ar];       b0.y = Bs[p][kk + krow + 1][wn*32 + ar];
            b1.x = Bs[p][kk + krow][wn*32 + 16 + ar];  b1.y = Bs[p][kk + krow + 1][wn*32 + 16 + ar];
            acc00 = WMMA_F32(a0, b0, acc00);
            acc01 = WMMA_F32(a0, b1, acc01);
            acc10 = WMMA_F32(a1, b0, acc10);
            acc11 = WMMA_F32(a1, b1, acc11);
        }

        if (has_next) {
            async_publish();
            __syncthreads();
            p ^= 1;
        }
    }

    const float g  = gamma[0];
    float* ob = out + (size_t)b * CCH * NSP;
    const int Mf = m0 + wm*32 + 8 * (lane >> 4);
    const int Nf = n0 + wn*32 + (lane & 15);
#pragma unroll
    for (int r = 0; r < 8; ++r) {
        const size_t i00 = (size_t)(Mf + r)      * NSP + Nf;
        const size_t i01 = (size_t)(Mf + r)      * NSP + Nf + 16;
        const size_t i10 = (size_t)(Mf + 16 + r) * NSP + Nf;
        const size_t i11 = (size_t)(Mf + 16 + r) * NSP + Nf + 16;
        ob[i00] = g * acc00[r] + xb[i00];
        ob[i01] = g * acc01[r] + xb[i01];
        ob[i10] = g * acc10[r] + xb[i10];
        ob[i11] = g * acc11[r] + xb[i11];
    }
}

// ---------------------------------------------------------------------------
extern "C" void kernel_launch(void* const* d_in, const int* in_sizes, int n_in,
                              void* d_out, int out_size, void* d_ws, size_t ws_size,
                              hipStream_t stream) {
    const float* x      = (const float*)d_in[0];
    const float* gamma  = (const float*)d_in[1];
    float*       out    = (float*)d_out;
    float*       energy = (float*)d_ws;   // needs 8*512*512*4 = 8 MB

    const dim3 blk(256);
    cam_energy <<<dim3(CCH / 128, CCH / 64, BATCH), blk, 0, stream>>>(x, energy);
    cam_softmax<<<dim3(BATCH * CCH),                blk, 0, stream>>>(energy);
    cam_out    <<<dim3(NSP / 128, CCH / 64, BATCH), blk, 0, stream>>>(x, energy, gamma, out);
}